// GCNConv_21466246546035
// MI455X (gfx1250) — compile-verified
//
#include <hip/hip_runtime.h>
#include <hip/hip_bf16.h>

typedef float v2f __attribute__((ext_vector_type(2)));
typedef float v8f __attribute__((ext_vector_type(8)));

#define DIM 128   // D_IN == D_OUT == 128
#define TM  16    // output rows per block
#define NTHREADS 256  // 8 waves; wave w computes output cols [16w, 16w+16)

// ---------------------------------------------------------------- zero fill
__global__ void zero_f32(float* __restrict__ p, int n) {
  int i = blockIdx.x * blockDim.x + threadIdx.x;
  if (i < n) p[i] = 0.f;
}

// ---------------------------------------------------------------- degrees
__global__ void degree_kernel(const long long* __restrict__ adj, int E,
                              float* __restrict__ sdeg, float* __restrict__ rdeg) {
  int e = blockIdx.x * blockDim.x + threadIdx.x;
  if (e < E) {
    atomicAdd(&sdeg[(int)adj[e]],     1.f);   // senders
    atomicAdd(&rdeg[(int)adj[E + e]], 1.f);   // receivers
  }
}

// ---------------------------------------------------------------- GEMM (WMMA f32)
// hs[m][n] = (sum_k x[m][k]*W[n][k] + b[n]) * rsqrt(max(sdeg[m],1))
// One block = 16 rows of x; wave w = 16 output cols. K=128 -> 32x V_WMMA_F32_16X16X4_F32.
__global__ __launch_bounds__(NTHREADS)
void gemm_scale_kernel(const float* __restrict__ x, const float* __restrict__ W,
                       const float* __restrict__ bias, const float* __restrict__ sdeg,
                       float* __restrict__ hs) {
  __shared__ float sx[TM * DIM];                 // 8 KB x-tile
  const int row0 = blockIdx.x * TM;
  const int tid  = threadIdx.x;

  // cooperative coalesced load of the 16x128 x tile (2048 floats / 256 thr)
  #pragma unroll
  for (int i = 0; i < (TM * DIM) / NTHREADS; ++i) {
    int idx = tid + i * NTHREADS;
    sx[idx] = x[(size_t)row0 * DIM + idx];
  }
  __syncthreads();

  const int wave = tid >> 5;
  const int lane = tid & 31;
  const int hi   = lane >> 4;     // 0: lanes 0-15 (K=0,1) | 1: lanes 16-31 (K=2,3)
  const int lo   = lane & 15;
  const int n0   = wave * 16;     // this wave's output-column tile

  v8f c = {0.f, 0.f, 0.f, 0.f, 0.f, 0.f, 0.f, 0.f};
  #pragma unroll
  for (int kk = 0; kk < DIM; kk += 4) {
    const int ka = kk + 2 * hi;
    // A frag: lane -> row M=lo, two consecutive K values (from LDS)
    v2f a = *(const v2f*)&sx[lo * DIM + ka];
    // B frag: B[k][n] = W[n][k]; lane -> col N=n0+lo, same two K values
    v2f bm = *(const v2f*)&W[(size_t)(n0 + lo) * DIM + ka];
    c = __builtin_amdgcn_wmma_f32_16x16x4_f32(false, a, false, bm,
                                              (short)0, c, false, false);
  }

  const float bv = bias[n0 + lo];
  // C/D layout: VGPR r -> M = r + 8*hi, N = lo
  #pragma unroll
  for (int r = 0; r < 8; ++r) {
    const int row = row0 + r + 8 * hi;
    const float sc = rsqrtf(fmaxf(sdeg[row], 1.f));
    hs[(size_t)row * DIM + n0 + lo] = (c[r] + bv) * sc;
  }
}

// ---------------------------------------------------------------- edge scatter-add
// 32 lanes per edge, float4 per lane: coalesced 512B row gather + fp32 atomics.
__global__ void scatter_kernel(const long long* __restrict__ adj, long long E,
                               const float* __restrict__ hs, float* __restrict__ out) {
  long long gid = (long long)blockIdx.x * blockDim.x + threadIdx.x;
  if (gid >= E * 32) return;
  const int e = (int)(gid >> 5);
  const int c = ((int)gid & 31) * 4;
  const int s = (int)adj[e];
  const int r = (int)adj[E + e];
  const float4 v = *(const float4*)&hs[(size_t)s * DIM + c];
  float* dst = &out[(size_t)r * DIM + c];
  atomicAdd(dst + 0, v.x);
  atomicAdd(dst + 1, v.y);
  atomicAdd(dst + 2, v.z);
  atomicAdd(dst + 3, v.w);
}

// ---------------------------------------------------------------- receiver scale + SiLU
__global__ void epilogue_kernel(float* __restrict__ out, const float* __restrict__ rdeg,
                                int total) {
  int i = blockIdx.x * blockDim.x + threadIdx.x;
  if (i < total) {
    const float sc = rsqrtf(fmaxf(rdeg[i / DIM], 1.f));
    const float v  = out[i] * sc;
    out[i] = v / (1.f + __expf(-v));    // silu(v) = v * sigmoid(v)
  }
}

// ---------------------------------------------------------------- launch
extern "C" void kernel_launch(void* const* d_in, const int* in_sizes, int n_in,
                              void* d_out, int out_size, void* d_ws, size_t ws_size,
                              hipStream_t stream) {
  const float*     x   = (const float*)d_in[0];      // [N,128] f32
  const long long* adj = (const long long*)d_in[1];  // [2,E] i64
  const float*     W   = (const float*)d_in[2];      // [128,128] f32
  const float*     b   = (const float*)d_in[3];      // [128] f32
  float* out = (float*)d_out;

  const int       N = in_sizes[0] / DIM;             // 100000 (multiple of 16)
  const long long E = in_sizes[1] / 2;               // 1,600,000

  // workspace: sdeg[N] | rdeg[N] | hs[N*128]  (~52 MB)
  float* sdeg = (float*)d_ws;
  float* rdeg = sdeg + N;
  float* hs   = rdeg + N;

  const int hElems = N * DIM;

  zero_f32<<<(2 * N + 255) / 256, 256, 0, stream>>>(sdeg, 2 * N);        // degrees
  zero_f32<<<(hElems + 255) / 256, 256, 0, stream>>>(out, hElems);       // scatter target

  degree_kernel<<<((int)E + 255) / 256, 256, 0, stream>>>(adj, (int)E, sdeg, rdeg);

  gemm_scale_kernel<<<N / TM, NTHREADS, 0, stream>>>(x, W, b, sdeg, hs);

  const long long sthreads = E * 32;
  scatter_kernel<<<(int)((sthreads + 255) / 256), 256, 0, stream>>>(adj, E, hs, out);

  epilogue_kernel<<<(hElems + 255) / 256, 256, 0, stream>>>(out, rdeg, hElems);

  // pass-through adj occupies the output tail (int64 bytes viewed as output elems)
  const long long tailElems = (long long)out_size - hElems;
  if (tailElems > 0) {
    hipMemcpyAsync((char*)d_out + (size_t)hElems * sizeof(float), d_in[1],
                   (size_t)tailElems * sizeof(float), hipMemcpyDeviceToDevice, stream);
  }
}